// GAT_12859132084769
// MI455X (gfx1250) — compile-verified
//
#include <hip/hip_runtime.h>
#include <stdint.h>

#define NEG_SLOPE 0.2f
#define GAT_EPS   1e-16f

typedef __attribute__((ext_vector_type(2))) float v2f;
typedef __attribute__((ext_vector_type(8))) float v8f;

// ---------- monotone float <-> uint key (for atomicMax-based segment max) ----
__device__ __forceinline__ unsigned enc_f32(float f) {
  unsigned u = __float_as_uint(f);
  return (u & 0x80000000u) ? ~u : (u | 0x80000000u);
}
__device__ __forceinline__ float dec_f32(unsigned k) {
  unsigned u = (k & 0x80000000u) ? (k & 0x7FFFFFFFu) : ~k;
  return __uint_as_float(u);
}

// ---------------------------- utility fill ----------------------------------
__global__ void fill_zero_u32(unsigned* __restrict__ p, long long n) {
  long long i = (long long)blockIdx.x * blockDim.x + threadIdx.x;
  if (i < n) p[i] = 0u;
}

// ------------- WMMA f32 strip GEMM: C = A(MxK) * B(KxN), f32 ---------------
// One wave32 per 16x(16*NT) output strip: A float2 loaded once per k-step and
// reused by NT V_WMMA_F32_16X16X4_F32 issues. All NT B-pairs are loaded into
// distinct registers BEFORE the WMMA burst so the backend can use staged
// s_wait_loadcnt thresholds instead of draining to 0 before each matrix op.
// VGPR layouts per CDNA5 ISA 7.12.2:
//   A 16x4 : lane(lo,hi): a0=A[row0+lo][k+2*hi], a1=A[row0+lo][k+2*hi+1]
//   B 4x16 : lane(lo,hi): b0=B[k+2*hi][c0+lo],  b1=B[k+2*hi+1][c0+lo]
//   C 16x16: acc[v] -> C[row0 + v + 8*hi][c0 + lo]
template <int NT>
__global__ void gemm_wmma_f32(const float* __restrict__ A,
                              const float* __restrict__ B,
                              float* __restrict__ C,
                              int M, int Nc, int K) {
  int wave    = blockIdx.x * (blockDim.x >> 5) + (threadIdx.x >> 5);
  int stripsN = Nc / (16 * NT);
  int tilesM  = M >> 4;                  // M = 50000 = 3125*16 (exact)
  if (wave >= tilesM * stripsN) return;  // whole-wave guard: EXEC all-1 inside
  int tm = wave / stripsN, ts = wave % stripsN;
  int row0 = tm << 4, col0 = ts * (16 * NT);
  int lane = threadIdx.x & 31;
  int lo = lane & 15, hi = lane >> 4;

  const float* arow  = A + (size_t)(row0 + lo) * K + 2 * hi;  // 8B aligned
  const float* bbase = B + (size_t)(2 * hi) * Nc + col0 + lo;

  v8f acc[NT];
#pragma unroll
  for (int t = 0; t < NT; t++) acc[t] = (v8f){};

  for (int k = 0; k < K; k += 4) {
    const float* b0 = bbase + (size_t)k * Nc;
    // --- issue phase: A (1x b64) + NT B pairs into distinct registers ---
    float2 af = *reinterpret_cast<const float2*>(arow + k);
    v2f bvs[NT];
#pragma unroll
    for (int t = 0; t < NT; t++) {
      bvs[t][0] = b0[t * 16];
      bvs[t][1] = b0[(size_t)Nc + t * 16];
    }
    v2f av; av[0] = af.x; av[1] = af.y;
    // --- consume phase: NT WMMAs (independent accumulator chains) ---
#pragma unroll
    for (int t = 0; t < NT; t++) {
      acc[t] = __builtin_amdgcn_wmma_f32_16x16x4_f32(
          /*neg_a=*/false, av, /*neg_b=*/false, bvs[t],
          /*c_mod=*/(short)0, acc[t], /*reuse_a=*/false, /*reuse_b=*/false);
    }
  }
#pragma unroll
  for (int t = 0; t < NT; t++)
#pragma unroll
    for (int v = 0; v < 8; v++)
      C[(size_t)(row0 + v + 8 * hi) * Nc + col0 + t * 16 + lo] = acc[t][v];
}

// ----------------- per-(node,head) attention projections --------------------
template <int H, int C>
__global__ void attn_proj(const float* __restrict__ h,
                          const float* __restrict__ att_src,
                          const float* __restrict__ att_dst,
                          float* __restrict__ a_src,
                          float* __restrict__ a_dst, int n_nodes) {
  int idx = blockIdx.x * blockDim.x + threadIdx.x;
  if (idx >= n_nodes * H) return;
  int n = idx / H, hh = idx % H;
  const float* hp  = h + (size_t)n * (H * C) + hh * C;
  const float* asp = att_src + hh * C;
  const float* adp = att_dst + hh * C;
  float s = 0.f, d = 0.f;
#pragma unroll
  for (int c = 0; c < C; c++) { float v = hp[c]; s += v * asp[c]; d += v * adp[c]; }
  a_src[idx] = s;
  a_dst[idx] = d;
}

// -------------------------- edge helpers ------------------------------------
__device__ __forceinline__ void edge_sd(const long long* __restrict__ esrc,
                                        const long long* __restrict__ edst,
                                        long long e, long long E,
                                        long long& s, long long& d) {
  if (e < E) { s = esrc[e]; d = edst[e]; }
  else       { s = e - E;   d = s;       }   // synthesized self-loop
}

template <int H>
__global__ void edge_max(const long long* __restrict__ esrc,
                         const long long* __restrict__ edst,
                         long long E, long long EE,
                         const float* __restrict__ a_src,
                         const float* __restrict__ a_dst,
                         unsigned* __restrict__ mkey) {
  long long idx = (long long)blockIdx.x * blockDim.x + threadIdx.x;
  if (idx >= EE * H) return;
  long long e = idx / H; int hh = (int)(idx % H);
  long long s, d; edge_sd(esrc, edst, e, E, s, d);
  float ev = a_src[s * H + hh] + a_dst[d * H + hh];
  ev = ev > 0.f ? ev : NEG_SLOPE * ev;
  atomicMax(mkey + d * H + hh, enc_f32(ev));
}

template <int H>
__global__ void edge_expsum(const long long* __restrict__ esrc,
                            const long long* __restrict__ edst,
                            long long E, long long EE,
                            const float* __restrict__ a_src,
                            const float* __restrict__ a_dst,
                            const unsigned* __restrict__ mkey,
                            float* __restrict__ ssum,
                            float* __restrict__ exbuf) {
  long long idx = (long long)blockIdx.x * blockDim.x + threadIdx.x;
  if (idx >= EE * H) return;
  long long e = idx / H; int hh = (int)(idx % H);
  long long s, d; edge_sd(esrc, edst, e, E, s, d);
  float ev = a_src[s * H + hh] + a_dst[d * H + hh];
  ev = ev > 0.f ? ev : NEG_SLOPE * ev;
  float mm = dec_f32(mkey[d * H + hh]);
  float ex = __expf(ev - mm);
  exbuf[idx] = ex;                       // exbuf indexed [e*H + hh] == idx
  atomicAdd(ssum + d * H + hh, ex);
}

// One thread per (edge, head): alpha computed once, channel groups unrolled.
template <int H, int C>
__global__ void edge_agg(const long long* __restrict__ esrc,
                         const long long* __restrict__ edst,
                         long long E, long long EE,
                         const float* __restrict__ exbuf,
                         const float* __restrict__ ssum,
                         const float* __restrict__ hfeat,
                         float* __restrict__ out) {
  constexpr int CG = C / 4;              // float4 channel groups
  long long idx = (long long)blockIdx.x * blockDim.x + threadIdx.x;
  if (idx >= EE * H) return;
  long long e = idx / H; int hh = (int)(idx % H);
  long long s, d; edge_sd(esrc, edst, e, E, s, d);
  float alpha = exbuf[idx] / (ssum[d * H + hh] + GAT_EPS);
  const float4* hp = reinterpret_cast<const float4*>(hfeat + (size_t)s * (H * C) + hh * C);
  float* op = out + (size_t)d * (H * C) + hh * C;
#pragma unroll
  for (int g = 0; g < CG; g++) {
    float4 hv = hp[g];
    atomicAdd(op + 4 * g + 0, hv.x * alpha);
    atomicAdd(op + 4 * g + 1, hv.y * alpha);
    atomicAdd(op + 4 * g + 2, hv.z * alpha);
    atomicAdd(op + 4 * g + 3, hv.w * alpha);
  }
}

__global__ void bias_act(float* __restrict__ out, const float* __restrict__ bias,
                         long long n_elems, int HC, int do_elu) {
  long long idx = (long long)blockIdx.x * blockDim.x + threadIdx.x;
  if (idx >= n_elems) return;
  float v = out[idx] + bias[idx % HC];
  if (do_elu) v = v > 0.f ? v : expm1f(v);
  out[idx] = v;
}

// ---------------------------------------------------------------------------
static inline unsigned nblk(long long n, int b) {
  return (unsigned)((n + b - 1) / b);
}

extern "C" void kernel_launch(void* const* d_in, const int* in_sizes, int n_in,
                              void* d_out, int out_size, void* d_ws, size_t ws_size,
                              hipStream_t stream) {
  const float*     x        = (const float*)d_in[0];
  const long long* eidx     = (const long long*)d_in[1];   // int64 per reference
  const float*     W1       = (const float*)d_in[2];
  const float*     att_src1 = (const float*)d_in[3];
  const float*     att_dst1 = (const float*)d_in[4];
  const float*     bias1    = (const float*)d_in[5];
  const float*     W2       = (const float*)d_in[6];
  const float*     att_src2 = (const float*)d_in[7];
  const float*     att_dst2 = (const float*)d_in[8];
  const float*     bias2    = (const float*)d_in[9];

  const int       Nn = in_sizes[0] / 128;           // 50000
  const long long Eg = (long long)in_sizes[1] / 2;  // 1,600,000
  const long long EE = Eg + Nn;                     // + self loops
  const long long* esrc = eidx;
  const long long* edst = eidx + Eg;

  // ---- workspace layout (all 16B-aligned chunks, ~94 MB total) ----
  char*  ws   = (char*)d_ws;
  size_t o    = 0;
  float* h1   = (float*)(ws + o); o += (size_t)Nn * 128 * 4;  // 25.6 MB
  float* out1 = (float*)(ws + o); o += (size_t)Nn * 128 * 4;  // 25.6 MB (layer2 input)
  float* h2   = (float*)(ws + o); o += (size_t)Nn * 64 * 4;   // 12.8 MB
  float*    a_s   = (float*)(ws + o);    o += (size_t)Nn * 4 * 4;
  float*    a_d   = (float*)(ws + o);    o += (size_t)Nn * 4 * 4;
  unsigned* mkey  = (unsigned*)(ws + o); o += (size_t)Nn * 4 * 4;
  float*    ssum  = (float*)(ws + o);    o += (size_t)Nn * 4 * 4;
  float*    exbuf = (float*)(ws + o);    o += (size_t)EE * 4 * 4; // 26.4 MB
  (void)ws_size; (void)n_in; (void)out_size;

  const int B = 256;

  // ================= Layer 1: H=4, C=32, HC=128 =================
  fill_zero_u32<<<nblk((long long)Nn * 128, B), B, 0, stream>>>((unsigned*)out1, (long long)Nn * 128);
  fill_zero_u32<<<nblk((long long)Nn * 4,   B), B, 0, stream>>>(mkey, (long long)Nn * 4);
  fill_zero_u32<<<nblk((long long)Nn * 4,   B), B, 0, stream>>>((unsigned*)ssum, (long long)Nn * 4);

  {
    long long waves = (long long)(Nn >> 4) * (128 / 64);     // 6250 strips
    gemm_wmma_f32<4><<<nblk(waves, 8), B, 0, stream>>>(x, W1, h1, Nn, 128, 128);
  }
  attn_proj<4, 32><<<nblk((long long)Nn * 4, B), B, 0, stream>>>(h1, att_src1, att_dst1, a_s, a_d, Nn);
  edge_max<4><<<nblk(EE * 4, B), B, 0, stream>>>(esrc, edst, Eg, EE, a_s, a_d, mkey);
  edge_expsum<4><<<nblk(EE * 4, B), B, 0, stream>>>(esrc, edst, Eg, EE, a_s, a_d, mkey, ssum, exbuf);
  edge_agg<4, 32><<<nblk(EE * 4, B), B, 0, stream>>>(esrc, edst, Eg, EE, exbuf, ssum, h1, out1);
  bias_act<<<nblk((long long)Nn * 128, B), B, 0, stream>>>(out1, bias1, (long long)Nn * 128, 128, 1);

  // ================= Layer 2: H=1, C=64, HC=64 ==================
  float* outf = (float*)d_out;
  fill_zero_u32<<<nblk((long long)Nn * 64, B), B, 0, stream>>>((unsigned*)outf, (long long)Nn * 64);
  fill_zero_u32<<<nblk((long long)Nn, B), B, 0, stream>>>(mkey, (long long)Nn);
  fill_zero_u32<<<nblk((long long)Nn, B), B, 0, stream>>>((unsigned*)ssum, (long long)Nn);

  {
    long long waves = (long long)(Nn >> 4) * (64 / 64);      // 3125 strips
    gemm_wmma_f32<4><<<nblk(waves, 8), B, 0, stream>>>(out1, W2, h2, Nn, 64, 128);
  }
  attn_proj<1, 64><<<nblk((long long)Nn, B), B, 0, stream>>>(h2, att_src2, att_dst2, a_s, a_d, Nn);
  edge_max<1><<<nblk(EE, B), B, 0, stream>>>(esrc, edst, Eg, EE, a_s, a_d, mkey);
  edge_expsum<1><<<nblk(EE, B), B, 0, stream>>>(esrc, edst, Eg, EE, a_s, a_d, mkey, ssum, exbuf);
  edge_agg<1, 64><<<nblk(EE, B), B, 0, stream>>>(esrc, edst, Eg, EE, exbuf, ssum, h2, outf);
  bias_act<<<nblk((long long)Nn * 64, B), B, 0, stream>>>(outf, bias2, (long long)Nn * 64, 64, 0);
}